// CoupledCLEFOModel_18880676233981
// MI455X (gfx1250) — compile-verified
//
#include <hip/hip_runtime.h>

typedef float v2f __attribute__((ext_vector_type(2)));
typedef float v8f __attribute__((ext_vector_type(8)));

#define N_DEP    32
#define N_INDEP  64
#define N_INTER  16
#define BATCH    65536
#define REG      1e-7f
#define NITER    16
#define WAVES    8
#define LDS_STRIDE 34   // 16x34 f32 tile per wave: conflict-free, keeps float2 reads 8B-aligned

// D = A(16x4, f32) x B(4x16, f32) + C(16x16, f32)
__device__ __forceinline__ v8f wmma_f32(v2f a, v2f b, v8f c) {
    return __builtin_amdgcn_wmma_f32_16x16x4_f32(
        /*neg_a=*/false, a, /*neg_b=*/false, b,
        /*c_mod=*/(short)0, c, /*reuse_a=*/false, /*reuse_b=*/false);
}

__global__ __launch_bounds__(256)
void clefo_jacobi_wmma(const float* __restrict__ X,   // [BATCH, 64]
                       const float* __restrict__ Z,   // [BATCH, 16]
                       const float* __restrict__ Ups, // [32, 1]
                       const float* __restrict__ B,   // [32, 64]
                       const float* __restrict__ Th,  // [32, 16]
                       const float* __restrict__ Ga,  // [32, 32]
                       const float* __restrict__ La,  // [32, 64]
                       float* __restrict__ Y)         // [BATCH, 32]
{
    __shared__ float lds[WAVES][16 * LDS_STRIDE];

    const int wave = threadIdx.x >> 5;
    const int lane = threadIdx.x & 31;
    const int half = lane >> 4;     // 0/1: which half of the wave
    const int lc   = lane & 15;
    const int tile = blockIdx.x * WAVES + wave;   // 16 batch rows per wave
    const int row0 = tile * 16;

    const float* Xt = X + (size_t)row0 * N_INDEP;
    const float* Zt = Z + (size_t)row0 * N_INTER;
    float* sl = &lds[wave][0];

    // ---- A fragments for the batch tile (shared by X@B^T and X@Lambda^T) ----
    // 32-bit A 16x4 layout: lane l -> row m=l%16, cols 4*k0 + 2*half + {0,1}
    v2f xa[16];
#pragma unroll
    for (int k0 = 0; k0 < 16; ++k0)
        xa[k0] = *(const v2f*)(Xt + lc * N_INDEP + 4 * k0 + 2 * half);
    v2f za[4];
#pragma unroll
    for (int k0 = 0; k0 < 4; ++k0)
        za[k0] = *(const v2f*)(Zt + lc * N_INTER + 4 * k0 + 2 * half);

    // ---- rhs = Ups + X@B^T + Z@Th^T ;  lam = X@La^T  (D layout: m=8*half+r, n=nt*16+lc) ----
    v8f rhs[2], lam[2];
#pragma unroll
    for (int nt = 0; nt < 2; ++nt) {
        rhs[nt] = (v8f)(0.0f);
        lam[nt] = (v8f)(0.0f);
        const int n = nt * 16 + lc;
#pragma unroll
        for (int k0 = 0; k0 < 16; ++k0) {
            // Bop[k][n] = W[n][k] : lane l -> n=l%16, k rows 4*k0+2*half+{0,1}
            v2f bb = *(const v2f*)(B  + n * N_INDEP + 4 * k0 + 2 * half);
            rhs[nt] = wmma_f32(xa[k0], bb, rhs[nt]);
            v2f lb = *(const v2f*)(La + n * N_INDEP + 4 * k0 + 2 * half);
            lam[nt] = wmma_f32(xa[k0], lb, lam[nt]);
        }
#pragma unroll
        for (int k0 = 0; k0 < 4; ++k0) {
            v2f tb = *(const v2f*)(Th + n * N_INTER + 4 * k0 + 2 * half);
            rhs[nt] = wmma_f32(za[k0], tb, rhs[nt]);
        }
        const float u = Ups[n];
#pragma unroll
        for (int r = 0; r < 8; ++r) rhs[nt][r] += u;
    }

    // ---- Jacobi diagonal: D = (1+reg) - lam ;  dinv = 1/D (elementwise in D layout) ----
    v8f dinv[2];
#pragma unroll
    for (int nt = 0; nt < 2; ++nt)
#pragma unroll
        for (int r = 0; r < 8; ++r)
            dinv[nt][r] = 1.0f / (1.0f + REG - lam[nt][r]);

    // ---- Gamma B-fragments (kept in registers across all iterations) ----
    // (Y @ Gamma^T)[m][n] = sum_k Y[m][k]*Gamma[n][k]  ->  Bop[k][n] = Gamma[n][k]
    v2f gb[2][8];
#pragma unroll
    for (int nt = 0; nt < 2; ++nt)
#pragma unroll
        for (int k0 = 0; k0 < 8; ++k0)
            gb[nt][k0] = *(const v2f*)(Ga + (nt * 16 + lc) * N_DEP + 4 * k0 + 2 * half);

    // ---- y0 = dinv * rhs ----
    v8f y[2];
#pragma unroll
    for (int nt = 0; nt < 2; ++nt)
#pragma unroll
        for (int r = 0; r < 8; ++r)
            y[nt][r] = rhs[nt][r] * dinv[nt][r];

    // ---- Jacobi iterations: y <- dinv * (rhs + Y @ Gamma^T) ----
    for (int it = 0; it < NITER; ++it) {
        // D layout -> LDS [m][k] tile (per-wave private region)
#pragma unroll
        for (int nt = 0; nt < 2; ++nt)
#pragma unroll
            for (int r = 0; r < 8; ++r)
                sl[(8 * half + r) * LDS_STRIDE + nt * 16 + lc] = y[nt][r];

        asm volatile("s_wait_dscnt 0" ::: "memory");  // DS in-order per wave; block reordering

        // Re-read as A-fragments: lane l -> m=l%16, k = 4*k0 + 2*half + {0,1}
        v2f ya[8];
#pragma unroll
        for (int k0 = 0; k0 < 8; ++k0)
            ya[k0] = *(const v2f*)(sl + lc * LDS_STRIDE + 4 * k0 + 2 * half);

        v8f t0 = (v8f)(0.0f), t1 = (v8f)(0.0f);
#pragma unroll
        for (int k0 = 0; k0 < 8; ++k0) {
            t0 = wmma_f32(ya[k0], gb[0][k0], t0);
            t1 = wmma_f32(ya[k0], gb[1][k0], t1);
        }
#pragma unroll
        for (int r = 0; r < 8; ++r) {
            y[0][r] = (rhs[0][r] + t0[r]) * dinv[0][r];
            y[1][r] = (rhs[1][r] + t1[r]) * dinv[1][r];
        }
    }

    // ---- store y: row = row0 + 8*half + r, col = nt*16 + lc (lanes 0..15 contiguous) ----
#pragma unroll
    for (int nt = 0; nt < 2; ++nt)
#pragma unroll
        for (int r = 0; r < 8; ++r)
            Y[(size_t)(row0 + 8 * half + r) * N_DEP + nt * 16 + lc] = y[nt][r];
}

extern "C" void kernel_launch(void* const* d_in, const int* in_sizes, int n_in,
                              void* d_out, int out_size, void* d_ws, size_t ws_size,
                              hipStream_t stream) {
    const float* X  = (const float*)d_in[0];
    const float* Z  = (const float*)d_in[1];
    const float* U  = (const float*)d_in[2];
    const float* B  = (const float*)d_in[3];
    const float* Th = (const float*)d_in[4];
    const float* Ga = (const float*)d_in[5];
    const float* La = (const float*)d_in[6];
    float* Y = (float*)d_out;

    dim3 block(32 * WAVES);                 // 256 threads = 8 wave32
    dim3 grid(BATCH / (16 * WAVES));        // 512 blocks, exact cover of 65536 rows
    hipLaunchKernelGGL(clefo_jacobi_wmma, grid, block, 0, stream,
                       X, Z, U, B, Th, Ga, La, Y);
}